// GraphMessagePassing_62989990363310
// MI455X (gfx1250) — compile-verified
//
#include <hip/hip_runtime.h>

typedef __attribute__((ext_vector_type(2))) float v2f;
typedef __attribute__((ext_vector_type(8))) float v8f;

#define N_ATOMS 20000
#define HIDDEN  128
#define BATCH   16
#define N_BONDS 60000
#define TILES   (N_ATOMS / 16)   // 1250 exact

// workspace layout in floats
#define WS_DEG  0
#define WS_M3   (N_ATOMS)            // 9 floats: M = W_msg @ W_upd (3x3 row-major)
#define WS_C2   (N_ATOMS + 9)        // 3 floats: c2 = b_msg @ W_upd
#define WS_AGG  (N_ATOMS + 16)       // BATCH*N_ATOMS*3 floats
#define WS_TOTAL (WS_AGG + BATCH * N_ATOMS * 3 + 4)  // +4 pad: allows ag[3] over-read

__global__ void k_zero(float* __restrict__ p, int n) {
    int i = blockIdx.x * blockDim.x + threadIdx.x;
    if (i < n) p[i] = 0.0f;
}

// Tiny precompute: M3 = W_msg(3x128) @ W_upd(128x3); c2 = b_msg(128) @ W_upd
__global__ void k_prep(const float* __restrict__ Wm, const float* __restrict__ bm,
                       const float* __restrict__ Wu, float* __restrict__ M3,
                       float* __restrict__ c2) {
    int t = threadIdx.x;
    if (t < 9) {
        int i = t / 3, j = t % 3;
        float acc = 0.0f;
        for (int k = 0; k < HIDDEN; ++k) acc += Wm[i * HIDDEN + k] * Wu[k * 3 + j];
        M3[i * 3 + j] = acc;
    } else if (t < 12) {
        int j = t - 9;
        float acc = 0.0f;
        for (int k = 0; k < HIDDEN; ++k) acc += bm[k] * Wu[k * 3 + j];
        c2[j] = acc;
    }
}

// deg[u] += 1, deg[v] += 1 per bond (both directions of the reference concat)
__global__ void k_deg(const int* __restrict__ bonds, float* __restrict__ deg) {
    int e = blockIdx.x * blockDim.x + threadIdx.x;
    if (e < N_BONDS) {
        int u = bonds[2 * e + 0];
        int v = bonds[2 * e + 1];
        atomicAdd(&deg[u], 1.0f);
        atomicAdd(&deg[v], 1.0f);
    }
}

// Collapsed scatter: agg[b][v] += h[b][u]; agg[b][u] += h[b][v]  (3 floats each)
__global__ void k_scatter(const float* __restrict__ h, const int* __restrict__ bonds,
                          float* __restrict__ agg) {
    long long tid = (long long)blockIdx.x * blockDim.x + threadIdx.x;
    if (tid >= (long long)N_BONDS * BATCH) return;
    int e = (int)(tid % N_BONDS);
    int b = (int)(tid / N_BONDS);
    int u = bonds[2 * e + 0];
    int v = bonds[2 * e + 1];
    const float* hu = h + ((long long)b * N_ATOMS + u) * 3;
    const float* hv = h + ((long long)b * N_ATOMS + v) * 3;
    float* au = agg + ((long long)b * N_ATOMS + u) * 3;
    float* av = agg + ((long long)b * N_ATOMS + v) * 3;
    float ux = hu[0], uy = hu[1], uz = hu[2];
    float vx = hv[0], vy = hv[1], vz = hv[2];
    atomicAdd(&av[0], ux); atomicAdd(&av[1], uy); atomicAdd(&av[2], uz);
    atomicAdd(&au[0], vx); atomicAdd(&au[1], vy); atomicAdd(&au[2], vz);
}

// Per-(batch, 16-node tile) update via V_WMMA_F32_16X16X4_F32:
//   D(16x16) = A(16x4) @ B(4x16) + C(16x16)
//   A rows  = s[node] = agg[node] * inv_deg[node], K-padded with 0
//   B       = M3 (3x3) in top-left, rest 0
//   C[m][n] = h[node m][n] + b_upd[n] + (deg>0 ? c2[n] : 0)  for n<3, else 0
// One wave per tile; wave32. All pre-WMMA predication is via cndmask-style
// selects on clamped-in-bounds addresses -> no divergence before the WMMA,
// EXEC is all-ones at issue. Single divergent region after for the stores.
__global__ void k_update(const float* __restrict__ hin, float* __restrict__ hout,
                         const float* __restrict__ agg, const float* __restrict__ deg,
                         const float* __restrict__ M3, const float* __restrict__ c2,
                         const float* __restrict__ b_upd) {
    int lane = threadIdx.x & 31;
    int w    = blockIdx.x * (blockDim.x >> 5) + (threadIdx.x >> 5);
    int b    = w / TILES;
    int tile = w % TILES;
    int base = tile * 16;
    int half = lane >> 4;          // ISA fp32 layout: lanes 0-15 = K0/K1 (A), rows 0-7 (C/D)
    int col  = lane & 15;
    int colc = col < 2 ? col : 2;  // clamped column index: always in-bounds
    bool pred = (col < 3);

    // ---- A (16x4 fp32, 2 VGPRs): lane<16 -> {s.x(K0), s.y(K1)}; lane>=16 -> {s.z(K2), 0(K3)}
    int nodeA  = base + col;
    float dA   = deg[nodeA];
    float invA = dA > 0.0f ? 1.0f / dA : 0.0f;
    const float* ag = agg + ((long long)b * N_ATOMS + nodeA) * 3;
    float a0 = ag[2 * half];       // half0: ag[0], half1: ag[2]
    float a1 = ag[2 * half + 1];   // half0: ag[1], half1: pad (discarded)
    v2f A;
    A.x = a0 * invA;
    A.y = half ? 0.0f : a1 * invA;

    // ---- B (4x16 fp32, 2 VGPRs): column = col; lane<16 -> rows K0,K1; lane>=16 -> K2, K3(=0)
    float m0 = M3[(half ? 6 : 0) + colc];  // half0: M[0][col], half1: M[2][col]
    float m1 = M3[3 + colc];               // M[1][col] (used by half0 only)
    v2f Bm;
    Bm.x = pred ? m0 : 0.0f;
    Bm.y = (pred && !half) ? m1 : 0.0f;

    // ---- C (16x16 fp32, 8 VGPRs): VGPR r = row (r + 8*half), column = col
    float bias = pred ? b_upd[colc] : 0.0f;
    float cc   = pred ? c2[colc]    : 0.0f;
    const float* hb = hin + ((long long)b * N_ATOMS + base + 8 * half) * 3 + colc;
    const float* db = deg + base + 8 * half;
    v8f C;
#pragma unroll
    for (int r = 0; r < 8; ++r) {
        float d  = db[r];
        float hv = hb[r * 3];                       // unconditional, clamped address
        C[r] = (pred ? hv : 0.0f) + bias + (d > 0.0f ? cc : 0.0f);  // bias,cc already 0 if !pred
    }

    v8f D = __builtin_amdgcn_wmma_f32_16x16x4_f32(
        /*neg_a=*/false, A, /*neg_b=*/false, Bm,
        /*c_mod=*/(short)0, C, /*reuse_a=*/false, /*reuse_b=*/false);

    if (pred) {
        float* ob = hout + ((long long)b * N_ATOMS + base + 8 * half) * 3 + col;
#pragma unroll
        for (int r = 0; r < 8; ++r) ob[r * 3] = D[r];
    }
}

extern "C" void kernel_launch(void* const* d_in, const int* in_sizes, int n_in,
                              void* d_out, int out_size, void* d_ws, size_t ws_size,
                              hipStream_t stream) {
    const float* positions = (const float*)d_in[0];
    const int*   bonds     = (const int*)d_in[1];
    const float* W_msg     = (const float*)d_in[2];
    const float* b_msg     = (const float*)d_in[3];
    const float* W_upd     = (const float*)d_in[4];
    const float* b_upd     = (const float*)d_in[5];
    float* out = (float*)d_out;
    float* ws  = (float*)d_ws;

    float* deg = ws + WS_DEG;
    float* M3  = ws + WS_M3;
    float* c2  = ws + WS_C2;
    float* agg = ws + WS_AGG;

    // zero deg + consts + agg (+pad), then precompute M3/c2 and degrees
    k_zero<<<(WS_TOTAL + 255) / 256, 256, 0, stream>>>(ws, WS_TOTAL);
    k_prep<<<1, 32, 0, stream>>>(W_msg, b_msg, W_upd, M3, c2);
    k_deg<<<(N_BONDS + 255) / 256, 256, 0, stream>>>(bonds, deg);

    const float* hin = positions;
    for (int it = 0; it < 2; ++it) {
        if (it > 0) {
            int na = BATCH * N_ATOMS * 3;
            k_zero<<<(na + 255) / 256, 256, 0, stream>>>(agg, na);
        }
        long long nscat = (long long)N_BONDS * BATCH;
        k_scatter<<<(int)((nscat + 255) / 256), 256, 0, stream>>>(hin, bonds, agg);
        // 20000 waves, 8 waves (256 threads) per block -> 2500 blocks exact
        k_update<<<(BATCH * TILES) / 8, 256, 0, stream>>>(hin, out, agg, deg, M3, c2, b_upd);
        hin = out; // iteration 2 updates d_out in place (per-lane read-then-write, no cross-lane hazard)
    }
}